// MeanAggregator_11433202942740
// MI455X (gfx1250) — compile-verified
//
#include <hip/hip_runtime.h>
#include <stdint.h>

typedef __attribute__((ext_vector_type(2))) float v2f;
typedef __attribute__((ext_vector_type(8))) float v8f;

#define D_FEAT 128
#define LDS_REC_CAP 8192u        // 32 KB of LDS record staging per block
#define PAD_REC 31u              // sentinel record: src=0, row=31 (never matches)

// ---------------------------------------------------------------------------
// Phase 1: per-tile edge counts (tile = dst >> 4) and per-node degree.
// ---------------------------------------------------------------------------
__global__ void count_kernel(const int* __restrict__ dst,
                             unsigned* __restrict__ counts,
                             unsigned* __restrict__ deg, int E) {
    int e = blockIdx.x * blockDim.x + threadIdx.x;
    if (e < E) {
        int d = dst[e];
        atomicAdd(&counts[((unsigned)d) >> 4], 1u);
        atomicAdd(&deg[d], 1u);
    }
}

// ---------------------------------------------------------------------------
// Phase 2: exclusive scan of tile counts, PADDED to multiples of 4 records
// (single block, nt <= 4096).
// ---------------------------------------------------------------------------
__global__ void scan_kernel(const unsigned* __restrict__ counts,
                            unsigned* __restrict__ offs, int nt) {
    __shared__ unsigned sums[1024];
    const int t = threadIdx.x;
    const int CH = 4;                       // 1024*4 = 4096 >= nt
    int start = t * CH;
    unsigned loc[CH];
    unsigned s = 0u;
    #pragma unroll
    for (int i = 0; i < CH; ++i) {
        int idx = start + i;
        unsigned v = (idx < nt) ? ((counts[idx] + 3u) & ~3u) : 0u;  // padded
        loc[i] = s;
        s += v;
    }
    sums[t] = s;
    __syncthreads();
    for (int off = 1; off < 1024; off <<= 1) {
        unsigned v = (t >= off) ? sums[t - off] : 0u;
        __syncthreads();
        sums[t] += v;
        __syncthreads();
    }
    unsigned base = (t > 0) ? sums[t - 1] : 0u;
    #pragma unroll
    for (int i = 0; i < CH; ++i) {
        int idx = start + i;
        if (idx < nt) offs[idx] = base + loc[i];
    }
    if (t == 1023) offs[nt] = sums[1023];   // padded total
}

// ---------------------------------------------------------------------------
// Phase 3a: scatter edges into per-tile bins. Record = (src << 5) | (dst & 15).
// ---------------------------------------------------------------------------
__global__ void scatter_kernel(const int* __restrict__ src,
                               const int* __restrict__ dst,
                               const unsigned* __restrict__ offs,
                               unsigned* __restrict__ cursors,
                               unsigned* __restrict__ bins, int E) {
    int e = blockIdx.x * blockDim.x + threadIdx.x;
    if (e < E) {
        int d = dst[e];
        unsigned t = ((unsigned)d) >> 4;
        unsigned pos = offs[t] + atomicAdd(&cursors[t], 1u);
        bins[pos] = (((unsigned)src[e]) << 5) | (((unsigned)d) & 15u);
    }
}

// ---------------------------------------------------------------------------
// Phase 3b: fill the <=3 padding slots per tile with sentinel records.
// ---------------------------------------------------------------------------
__global__ void pad_kernel(const unsigned* __restrict__ counts,
                           const unsigned* __restrict__ offs,
                           unsigned* __restrict__ bins, int nt) {
    int t = blockIdx.x * blockDim.x + threadIdx.x;
    if (t < nt) {
        unsigned c = counts[t];
        unsigned b = offs[t];
        unsigned p = offs[t + 1] - b;       // padded count
        for (unsigned i = c; i < p; ++i) bins[b + i] = PAD_REC;
    }
}

// ---------------------------------------------------------------------------
// Phase 4: WMMA SpMM. Block = one dst-tile (16 nodes), 8 waves x 16 features.
//   A (16x4 one-hot, v2f):  lane half hf, component j -> K = 2*hf + j.
//   B (4x16 gathered rows): VGPR0 = {K0 row | K2 row}, VGPR1 = {K1 | K3}.
//   C/D (16x16 f32, v8f):   VGPR p, half hf -> M = p + 8*hf, N = lane&15.
// Records staged to LDS once per block via CDNA5 async-to-LDS (ASYNCcnt),
// hot loop reads ds_load_b64; unroll 4 keeps ~8 gathers + 4 WMMAs in flight.
// All loop bounds are block-uniform so EXEC is all-1s at every WMMA.
// ---------------------------------------------------------------------------
__global__ void __launch_bounds__(256, 4)
spmm_wmma_kernel(const float* __restrict__ h,
                 const unsigned* __restrict__ offs,
                 const unsigned* __restrict__ bins,
                 const unsigned* __restrict__ deg,
                 float* __restrict__ out, int n_dst) {
    __shared__ unsigned sbin[LDS_REC_CAP];

    const int tile = blockIdx.x;
    const int wave = threadIdx.x >> 5;
    const int lane = threadIdx.x & 31;
    const int hf   = lane >> 4;          // lane half -> K pair (2*hf, 2*hf+1)
    const int m    = lane & 15;          // A row / C column
    const unsigned n = (unsigned)(m + (wave << 4));  // feature column

    const unsigned base = offs[tile];
    const unsigned pcnt = offs[tile + 1] - base;   // multiple of 4
    const bool useLds = (pcnt <= LDS_REC_CAP);

    v8f acc = {0.f, 0.f, 0.f, 0.f, 0.f, 0.f, 0.f, 0.f};

    if (useLds) {
        // ---- async stage records to LDS (wave 0 only), 512B per instruction
        if (wave == 0) {
            const unsigned nbytes = pcnt * 4u;              // multiple of 16
            unsigned ldsBase = (unsigned)(uintptr_t)(&sbin[0]);
            for (unsigned off = (unsigned)lane * 16u; off < nbytes; off += 512u) {
                unsigned ldsOff = ldsBase + off;
                unsigned gOff   = base * 4u + off;
                asm volatile("global_load_async_to_lds_b128 %0, %1, %2"
                             :: "v"(ldsOff), "v"(gOff), "s"(bins)
                             : "memory");
            }
            asm volatile("s_wait_asynccnt 0x0" ::: "memory");
        }
        __syncthreads();

        #pragma unroll 4
        for (unsigned g = 0; g < pcnt; g += 4u) {
            unsigned i0 = g + (unsigned)(hf << 1);
            uint2 rr = *reinterpret_cast<const uint2*>(&sbin[i0]);  // ds_load_b64
            float ax = ((rr.x & 31u) == (unsigned)m) ? 1.0f : 0.0f;
            float ay = ((rr.y & 31u) == (unsigned)m) ? 1.0f : 0.0f;
            v2f a = {ax, ay};
            // 32-bit gather offsets -> saddr + voffset addressing
            unsigned ix = (rr.x >> 5) * (unsigned)D_FEAT + n;
            unsigned iy = (rr.y >> 5) * (unsigned)D_FEAT + n;
            v2f b = {h[ix], h[iy]};
            acc = __builtin_amdgcn_wmma_f32_16x16x4_f32(
                false, a, false, b, (short)0, acc, false, false);
        }
    } else {
        // overflow path: records straight from global (block-uniform branch)
        #pragma unroll 4
        for (unsigned g = 0; g < pcnt; g += 4u) {
            unsigned i0 = g + (unsigned)(hf << 1);
            uint2 rr = *reinterpret_cast<const uint2*>(bins + base + i0);
            float ax = ((rr.x & 31u) == (unsigned)m) ? 1.0f : 0.0f;
            float ay = ((rr.y & 31u) == (unsigned)m) ? 1.0f : 0.0f;
            v2f a = {ax, ay};
            unsigned ix = (rr.x >> 5) * (unsigned)D_FEAT + n;
            unsigned iy = (rr.y >> 5) * (unsigned)D_FEAT + n;
            v2f b = {h[ix], h[iy]};
            acc = __builtin_amdgcn_wmma_f32_16x16x4_f32(
                false, a, false, b, (short)0, acc, false, false);
        }
    }

    #pragma unroll
    for (int p = 0; p < 8; ++p) {
        int M = p + (hf << 3);
        int node = (tile << 4) + M;
        if (node < n_dst) {
            float dv = (float)deg[node];
            out[(size_t)node * D_FEAT + n] = acc[p] / fmaxf(dv, 1.0f);
        }
    }
}

// ---------------------------------------------------------------------------
// Fallback path (used if d_ws is too small for the binning pipeline or nt > 4096).
// ---------------------------------------------------------------------------
__global__ void fb_scatter(const float* __restrict__ h,
                           const int* __restrict__ src,
                           const int* __restrict__ dst,
                           float* __restrict__ out, int E) {
    long long t = (long long)blockIdx.x * blockDim.x + threadIdx.x;
    long long e = t >> 5;
    int lane = (int)(t & 31);
    if (e < E) {
        const float* row = h + (size_t)src[e] * D_FEAT + lane * 4;
        float4 v = *(const float4*)row;
        float* o = out + (size_t)dst[e] * D_FEAT + lane * 4;
        atomicAdd(o + 0, v.x);
        atomicAdd(o + 1, v.y);
        atomicAdd(o + 2, v.z);
        atomicAdd(o + 3, v.w);
    }
}

__global__ void fb_deg(const int* __restrict__ dst, unsigned* __restrict__ deg, int E) {
    int e = blockIdx.x * blockDim.x + threadIdx.x;
    if (e < E) atomicAdd(&deg[dst[e]], 1u);
}

__global__ void fb_final(float* __restrict__ out, const unsigned* __restrict__ deg, int total) {
    int i = blockIdx.x * blockDim.x + threadIdx.x;
    if (i < total) {
        float dv = (float)deg[i >> 7];   // D_FEAT = 128
        out[i] = out[i] / fmaxf(dv, 1.0f);
    }
}

// ---------------------------------------------------------------------------
extern "C" void kernel_launch(void* const* d_in, const int* in_sizes, int n_in,
                              void* d_out, int out_size, void* d_ws, size_t ws_size,
                              hipStream_t stream) {
    const float* h   = (const float*)d_in[0];
    const int*   src = (const int*)d_in[1];
    const int*   dst = (const int*)d_in[2];
    float*       out = (float*)d_out;

    const int E     = in_sizes[1];
    const int n_dst = out_size / D_FEAT;
    const int nt    = (n_dst + 15) / 16;
    const int TB    = 256;

    unsigned* w = (unsigned*)d_ws;
    // bins need E + 3*nt entries (padding), plus counts/cursors/offs/deg
    size_t need = ((size_t)3 * nt + 1 + (size_t)n_dst + (size_t)E + 3 * (size_t)nt)
                  * sizeof(unsigned);

    if (ws_size >= need && nt <= 4096) {
        unsigned* counts  = w;                       // [nt]
        unsigned* cursors = w + nt;                  // [nt]
        unsigned* offs    = w + 2 * nt;              // [nt+1]
        unsigned* deg     = w + 3 * nt + 1;          // [n_dst]
        unsigned* bins    = deg + n_dst;             // [E + 3*nt]

        hipMemsetAsync(counts, 0, (size_t)2 * nt * sizeof(unsigned), stream);
        hipMemsetAsync(deg, 0, (size_t)n_dst * sizeof(unsigned), stream);

        count_kernel<<<(E + TB - 1) / TB, TB, 0, stream>>>(dst, counts, deg, E);
        scan_kernel<<<1, 1024, 0, stream>>>(counts, offs, nt);
        scatter_kernel<<<(E + TB - 1) / TB, TB, 0, stream>>>(src, dst, offs, cursors, bins, E);
        pad_kernel<<<(nt + TB - 1) / TB, TB, 0, stream>>>(counts, offs, bins, nt);
        spmm_wmma_kernel<<<nt, 256, 0, stream>>>(h, offs, bins, deg, out, n_dst);
    } else {
        // Atomic-scatter fallback
        hipMemsetAsync(out, 0, (size_t)out_size * sizeof(float), stream);
        unsigned* deg = w;                           // requires n_dst*4 bytes
        hipMemsetAsync(deg, 0, (size_t)n_dst * sizeof(unsigned), stream);

        long long threads = (long long)E * 32;
        fb_scatter<<<(unsigned)((threads + TB - 1) / TB), TB, 0, stream>>>(h, src, dst, out, E);
        fb_deg<<<(E + TB - 1) / TB, TB, 0, stream>>>(dst, deg, E);
        fb_final<<<(out_size + TB - 1) / TB, TB, 0, stream>>>(out, deg, out_size);
    }
}